// LCoDBase_73108933313174
// MI455X (gfx1250) — compile-verified
//
#include <hip/hip_runtime.h>
#include <stddef.h>

typedef __attribute__((ext_vector_type(2))) float v2f;
typedef __attribute__((ext_vector_type(4))) float v4f;
typedef __attribute__((ext_vector_type(8))) float v8f;

#define HD 4096   // hidden dim (M of GEMM, rows of Z/B/S)
#define KD 1024   // x_dim (K of GEMM)
#define ND 2048   // batch columns (N of GEMM)

#define BM 128
#define BN 128
#define BK 16
#define PAD 4     // keeps 16B alignment of rows (stride 20 floats) and spreads banks

// ---------------------------------------------------------------------------
// Kernel 1: B = We @ x  using V_WMMA_F32_16X16X4_F32
//   We: (HD, KD) row-major, x: (KD, ND) row-major, B: (HD, ND) row-major
// ---------------------------------------------------------------------------
__global__ __launch_bounds__(256)
void lcod_gemm(const float* __restrict__ We, const float* __restrict__ x,
               float* __restrict__ Bmat)
{
    __shared__ float As[2][BM][BK + PAD];   // [m][k]
    __shared__ float Bs[2][BN][BK + PAD];   // transposed: [n][k]

    const int t    = threadIdx.x;
    const int lane = t & 31;
    const int wave = t >> 5;
    const int m0   = blockIdx.y * BM;
    const int n0   = blockIdx.x * BN;

    // 8 waves arranged 4(M) x 2(N); each wave owns a 32x64 tile = 2x4 subtiles
    const int mWave = (wave >> 1) * 32;
    const int nWave = (wave & 1) * 64;

    const int l15   = lane & 15;
    const int lhalf = lane >> 4;   // 0 or 1

    v8f acc[2][4];
    const v8f vzero = {0.f,0.f,0.f,0.f,0.f,0.f,0.f,0.f};
    for (int i = 0; i < 2; ++i)
        for (int j = 0; j < 4; ++j) acc[i][j] = vzero;

    auto stage = [&](int kt, int buf) {
        const int k0 = kt * BK;
        // --- A tile: 128 x 16 floats; each thread loads 8 floats (2 x b128)
        {
            const int m  = t >> 1;           // 0..127
            const int kq = (t & 1) * 8;      // 0 or 8
            const float* g = We + (size_t)(m0 + m) * KD + (k0 + kq);
            v4f a0 = *(const v4f*)(g);
            v4f a1 = *(const v4f*)(g + 4);
            *(v4f*)&As[buf][m][kq]     = a0;
            *(v4f*)&As[buf][m][kq + 4] = a1;
        }
        // --- x tile: 16 x 128 floats; load row-wise, store transposed [n][k]
        {
            const int k  = t >> 4;           // 0..15
            const int cb = (t & 15) * 8;     // column base, 8 floats per thread
            const float* g = x + (size_t)(k0 + k) * ND + (n0 + cb);
            v4f b0 = *(const v4f*)(g);
            v4f b1 = *(const v4f*)(g + 4);
            Bs[buf][cb + 0][k] = b0.x; Bs[buf][cb + 1][k] = b0.y;
            Bs[buf][cb + 2][k] = b0.z; Bs[buf][cb + 3][k] = b0.w;
            Bs[buf][cb + 4][k] = b1.x; Bs[buf][cb + 5][k] = b1.y;
            Bs[buf][cb + 6][k] = b1.z; Bs[buf][cb + 7][k] = b1.w;
        }
    };

    const int NT = KD / BK;   // 64 K-steps
    stage(0, 0);
    __syncthreads();

    for (int kt = 0; kt < NT; ++kt) {
        const int cur = kt & 1;
        if (kt + 1 < NT) stage(kt + 1, cur ^ 1);

        // 4 WMMA K-substeps of K=4 each
        #pragma unroll
        for (int ks = 0; ks < 4; ++ks) {
            const int kk = ks * 4 + 2 * lhalf;   // lane holds K = kk, kk+1
            v2f afrag[2], bfrag[4];
            #pragma unroll
            for (int i = 0; i < 2; ++i)
                afrag[i] = *(const v2f*)&As[cur][mWave + 16 * i + l15][kk];
            #pragma unroll
            for (int j = 0; j < 4; ++j)
                bfrag[j] = *(const v2f*)&Bs[cur][nWave + 16 * j + l15][kk];
            #pragma unroll
            for (int i = 0; i < 2; ++i)
                #pragma unroll
                for (int j = 0; j < 4; ++j)
                    acc[i][j] = __builtin_amdgcn_wmma_f32_16x16x4_f32(
                        false, afrag[i], false, bfrag[j],
                        (short)0, acc[i][j], false, false);
        }
        __syncthreads();
    }

    // Epilogue: C/D layout: VGPR v -> M = v + 8*(lane>>4), N = lane&15
    #pragma unroll
    for (int i = 0; i < 2; ++i) {
        #pragma unroll
        for (int j = 0; j < 4; ++j) {
            const int n = n0 + nWave + 16 * j + l15;
            const int mb = m0 + mWave + 16 * i + 8 * lhalf;
            #pragma unroll
            for (int v = 0; v < 8; ++v)
                Bmat[(size_t)(mb + v) * ND + n] = acc[i][j][v];
        }
    }
}

// ---------------------------------------------------------------------------
// Kernel 2: per-column argmax_k |Z - shrink(B,theta)|, emit k[n], e[n]
//   16 columns per block, 1024 threads (64 row-threads per column)
// ---------------------------------------------------------------------------
__global__ __launch_bounds__(1024)
void lcod_select(const float* __restrict__ Z, const float* __restrict__ Bmat,
                 const float* __restrict__ theta_p,
                 int* __restrict__ kOut, float* __restrict__ eOut)
{
    __shared__ float sv[1024];
    __shared__ int   si[1024];

    const float theta = theta_p[0];
    const int t  = threadIdx.x;
    const int c  = t & 15;
    const int rt = t >> 4;
    const int n  = blockIdx.x * 16 + c;

    float best  = -1.0f;
    int   bestk = HD;
    for (int h = rt; h < HD; h += 64) {
        const float b  = Bmat[(size_t)h * ND + n];
        const float zb = fmaxf(b - theta, 0.0f) - fmaxf(-b - theta, 0.0f);
        const float d  = fabsf(Z[(size_t)h * ND + n] - zb);
        if (d > best) { best = d; bestk = h; }   // ascending h => first max kept
    }
    sv[t] = best;
    si[t] = bestk;
    __syncthreads();

    for (int s = 512; s >= 16; s >>= 1) {
        if (t < s) {
            const float v2 = sv[t + s];
            const int   i2 = si[t + s];
            if (v2 > sv[t] || (v2 == sv[t] && i2 < si[t])) { sv[t] = v2; si[t] = i2; }
        }
        __syncthreads();
    }

    if (rt == 0) {
        const int k   = si[c];
        const float b = Bmat[(size_t)k * ND + n];
        const float zb = fmaxf(b - theta, 0.0f) - fmaxf(-b - theta, 0.0f);
        kOut[n] = k;
        eOut[n] = zb - Z[(size_t)k * ND + n];
    }
}

// ---------------------------------------------------------------------------
// Kernel 3: out[0] = Z_new, out[1] = B_new
//   Z_new[h,n] = Z[h,n] + (h==k_n ? e_n : 0)
//   B_new[h,n] = B[h,n] + S[h,k_n] * e_n
// ---------------------------------------------------------------------------
__global__ __launch_bounds__(256)
void lcod_update(const float* __restrict__ Z, const float* __restrict__ Bmat,
                 const float* __restrict__ S,
                 const int* __restrict__ kIdx, const float* __restrict__ eVec,
                 float* __restrict__ out)
{
    const int n = blockIdx.x * 256 + threadIdx.x;
    const int h = blockIdx.y;
    const int k   = kIdx[n];
    const float e = eVec[n];

    const size_t idx = (size_t)h * ND + n;
    const float z = Z[idx];
    const float b = Bmat[idx] + S[(size_t)h * HD + k] * e;

    out[idx] = (h == k) ? (z + e) : z;
    out[(size_t)HD * ND + idx] = b;
}

// ---------------------------------------------------------------------------
extern "C" void kernel_launch(void* const* d_in, const int* in_sizes, int n_in,
                              void* d_out, int out_size, void* d_ws, size_t ws_size,
                              hipStream_t stream)
{
    (void)in_sizes; (void)n_in; (void)out_size; (void)ws_size;
    const float* x     = (const float*)d_in[0];   // (KD, ND)
    const float* Z     = (const float*)d_in[1];   // (HD, ND)
    const float* We    = (const float*)d_in[2];   // (HD, KD)
    const float* S     = (const float*)d_in[3];   // (HD, HD)
    const float* theta = (const float*)d_in[4];   // (1,)
    float* out = (float*)d_out;                   // (2, HD, ND)

    // workspace layout
    float* Bmat = (float*)d_ws;                                       // HD*ND f32
    int*   kIdx = (int*)((char*)d_ws + (size_t)HD * ND * sizeof(float));
    float* eVec = (float*)((char*)kIdx + (size_t)ND * sizeof(int));

    dim3 gGemm(ND / BN, HD / BM);           // (16, 32)
    lcod_gemm<<<gGemm, 256, 0, stream>>>(We, x, Bmat);

    lcod_select<<<ND / 16, 1024, 0, stream>>>(Z, Bmat, theta, kIdx, eVec);

    dim3 gUpd(ND / 256, HD);                // (8, 4096)
    lcod_update<<<gUpd, 256, 0, stream>>>(Z, Bmat, S, kIdx, eVec, out);
}